// MyRGCNConvOpt2_35536559407734
// MI455X (gfx1250) — compile-verified
//
#include <hip/hip_runtime.h>

#define NUM_REL      8
#define NUM_NODE     100000
#define T_NODE       50000
#define E_REL        800000
#define C            128     // IN_C == HID_C
#define TILES_PER_WG 5       // 3125 tiles / 5 = 625 workgroups per relation

typedef __attribute__((ext_vector_type(2))) float v2f;
typedef __attribute__((ext_vector_type(8))) float v8f;

// One workgroup = one relation r, TILES_PER_WG consecutive tiles of 16 target
// rows. 256 threads = 8 wave32s. B fragments (wave's 16 columns of W_r, all
// 128 K) are loaded into registers ONCE per workgroup and reused across tiles,
// cutting W L2 traffic 5x vs. per-tile reloads.
//  Phase 1 (per tile): gather+segment-sum 16 rows of agg (16x128 f32) into LDS.
//  Phase 2 (per tile): wave w computes the 16x16 output N-tile w with
//                      32x V_WMMA_F32_16X16X4_F32 (A from LDS, B from regs).
//  Phase 3 (per tile): scatter-add rows into out[sub_target] with f32 atomics.
__global__ __launch_bounds__(256) void rgcn_fused_kernel(
    const float* __restrict__ x,           // [NUM_NODE][C]
    const float* __restrict__ rel_weight,  // [NUM_REL][C][C]  (K x N row-major)
    const int*   __restrict__ sub_ptr,     // [NUM_REL][T_NODE+1]
    const int*   __restrict__ sub_idx,     // [NUM_REL][E_REL]
    const int*   __restrict__ sub_target,  // [NUM_REL][T_NODE]
    float*       __restrict__ out)         // [NUM_NODE][C]
{
  __shared__ float4 agg4[16 * (C / 4)];    // 16 rows x 128 f32 = 8 KB
  float* aggf = (float*)agg4;

  const int r    = blockIdx.y;
  const int tid  = threadIdx.x;
  const int wave = tid >> 5;
  const int lane = tid & 31;
  const int ml   = lane & 15;       // A: M index / B,D: N index within tile
  const int hi   = lane >> 4;       // lane group: K offset 0 or 2
  const int n    = wave * 16 + ml;  // this wave's global output column

  const int*    ptr = sub_ptr + r * (T_NODE + 1);
  const int*    idx = sub_idx + (size_t)r * E_REL;
  const float4* x4  = (const float4*)x;

  // ---- One-time B-fragment preload: W_r columns [16w,16w+16), all K -------
  // Lane holds W[ka][n], W[ka+1][n] for ka = 4*kk + 2*hi  (32 x v2f = 64 VGPR)
  const float* W = rel_weight + (size_t)r * C * C;
  v2f b[C / 4];
  #pragma unroll
  for (int kk = 0; kk < C / 4; ++kk) {
    const int ka = kk * 4 + 2 * hi;
    b[kk].x = W[(size_t)ka * C + n];
    b[kk].y = W[(size_t)(ka + 1) * C + n];
  }

  for (int tt = 0; tt < TILES_PER_WG; ++tt) {
    const int t0 = (blockIdx.x * TILES_PER_WG + tt) * 16;

    // ---------------- Phase 1: aggregation (gather + segment sum) -----------
    // Wave handles rows tl = 2*wave, 2*wave+1; each lane owns 4 channels.
    for (int rr = 0; rr < 2; ++rr) {
      const int tl = wave * 2 + rr;
      const int t  = t0 + tl;
      // searchsorted-with-clip semantics: row 0 starts at edge 0,
      // row T-1 extends to E (matches jnp.clip(searchsorted-1, 0, T-1)).
      int e0 = (t == 0) ? 0 : ptr[t];
      int e1 = (t == T_NODE - 1) ? E_REL : ptr[t + 1];
      e0 = __builtin_amdgcn_readfirstlane(e0);
      e1 = __builtin_amdgcn_readfirstlane(e1);
      float4 acc; acc.x = 0.f; acc.y = 0.f; acc.z = 0.f; acc.w = 0.f;
      for (int e = e0; e < e1; ++e) {
        const int nbr = __builtin_amdgcn_readfirstlane(idx[e]);  // wave-uniform
        const float4 v = x4[(size_t)nbr * (C / 4) + lane];       // 512B row read
        acc.x += v.x; acc.y += v.y; acc.z += v.z; acc.w += v.w;
      }
      agg4[tl * (C / 4) + lane] = acc;
    }
    __syncthreads();

    // ---------------- Phase 2: 16x128 @ 128x128 tile GEMM via f32 WMMA ------
    v8f cacc = {};                    // 16x16 f32 accumulator tile
    #pragma unroll
    for (int kk = 0; kk < C / 4; ++kk) {
      // A fragment (16x4): lane holds agg[M][ka], agg[M][ka+1]
      v2f a = *(const v2f*)(aggf + ml * C + kk * 4 + 2 * hi);
      cacc = __builtin_amdgcn_wmma_f32_16x16x4_f32(
          /*neg_a=*/false, a, /*neg_b=*/false, b[kk],
          /*c_mod=*/(short)0, cacc, /*reuse_a=*/false, /*reuse_b=*/false);
    }

    // ---------------- Phase 3: scatter-add rows into out[target] ------------
    // D layout: VGPR i -> (M = i + 8*hi, N = ml) for lanes split at 16.
    const int* tgt = sub_target + (size_t)r * T_NODE + t0;
    #pragma unroll
    for (int i = 0; i < 8; ++i) {
      const int node = tgt[i + 8 * hi];
      unsafeAtomicAdd(out + (size_t)node * C + n, cacc[i]);
    }
    __syncthreads();   // protect LDS agg before next tile's phase 1 writes
  }
}

extern "C" void kernel_launch(void* const* d_in, const int* in_sizes, int n_in,
                              void* d_out, int out_size, void* d_ws, size_t ws_size,
                              hipStream_t stream) {
  const float* x   = (const float*)d_in[0];
  const float* w   = (const float*)d_in[1];
  const int*   ptr = (const int*)d_in[2];
  const int*   idx = (const int*)d_in[3];
  const int*   tgt = (const int*)d_in[4];
  float*       out = (float*)d_out;

  // Output is accumulated with atomics -> must start from zero every call.
  hipMemsetAsync(out, 0, (size_t)NUM_NODE * C * sizeof(float), stream);

  dim3 grid(T_NODE / 16 / TILES_PER_WG, NUM_REL);   // 625 x 8 workgroups
  rgcn_fused_kernel<<<grid, 256, 0, stream>>>(x, w, ptr, idx, tgt, out);
}